// LoCoScaleAdaptive_45234595561959
// MI455X (gfx1250) — compile-verified
//
#include <hip/hip_runtime.h>
#include <math.h>

typedef float v2f __attribute__((ext_vector_type(2)));
typedef float v8f __attribute__((ext_vector_type(8)));

#define BB 32
#define NN 1024
#define DIN 64
#define HH 128
#define OO 8
#define LL 3
#define KK 24

__device__ __forceinline__ float warp_sum(float v) {
    #pragma unroll
    for (int off = 16; off > 0; off >>= 1) v += __shfl_xor(v, off, 32);
    return v;
}

__device__ __forceinline__ float gelu_f(float x) {
    return 0.5f * x * (1.0f + erff(x * 0.70710678118654752440f));
}

__device__ __forceinline__ float sigmoid_f(float x) {
    return 1.0f / (1.0f + expf(-x));
}

// ---------------------------------------------------------------------------
// Kernel 1: feature encoder.  16 tokens per 1-wave block.
//   h1 = gelu(ln(x @ fe_w1^T + b1)); spatial = ln(h1 @ fe_w2^T + b2)
// Also writes fs2[token] = sum(spatial^2).
// ---------------------------------------------------------------------------
__global__ void fe_kernel(const float* __restrict__ x,
                          const float* __restrict__ w1, const float* __restrict__ b1,
                          const float* __restrict__ g1, const float* __restrict__ bb1,
                          const float* __restrict__ w2, const float* __restrict__ b2,
                          const float* __restrict__ g2, const float* __restrict__ bb2,
                          float* __restrict__ spatial, float* __restrict__ fs2) {
    __shared__ float xt[16 * 64];
    __shared__ float h1[16 * 128];
    __shared__ float sp[16 * 128];
    const int tok0 = blockIdx.x * 16;
    const int l = threadIdx.x;
    for (int i = l; i < 16 * 64; i += 32)
        xt[i] = x[(size_t)(tok0 + (i >> 6)) * 64 + (i & 63)];
    __syncthreads();

    const int mr = l & 15, kh = l >> 4;
    // GEMM1: (16x64) @ (64x128)
    for (int nt = 0; nt < 8; ++nt) {
        v8f c = {};
        const int col = nt * 16 + mr;
        #pragma unroll
        for (int kb = 0; kb < 64; kb += 4) {
            v2f a, bf;
            a.x = xt[mr * 64 + kb + 2 * kh + 0];
            a.y = xt[mr * 64 + kb + 2 * kh + 1];
            bf.x = w1[col * 64 + kb + 2 * kh + 0];
            bf.y = w1[col * 64 + kb + 2 * kh + 1];
            c = __builtin_amdgcn_wmma_f32_16x16x4_f32(false, a, false, bf, (short)0, c, false, false);
        }
        #pragma unroll
        for (int r = 0; r < 8; ++r)
            h1[(r + 8 * kh) * 128 + col] = c[r] + b1[col];
    }
    __syncthreads();
    // LN + GELU per token (each lane owns columns l, l+32, l+64, l+96)
    for (int m = 0; m < 16; ++m) {
        float v[4];
        #pragma unroll
        for (int q = 0; q < 4; ++q) v[q] = h1[m * 128 + l + 32 * q];
        float mean = warp_sum(v[0] + v[1] + v[2] + v[3]) * (1.0f / 128.0f);
        float sq = 0.f;
        #pragma unroll
        for (int q = 0; q < 4; ++q) { float d = v[q] - mean; sq += d * d; }
        float inv = rsqrtf(warp_sum(sq) * (1.0f / 128.0f) + 1e-5f);
        #pragma unroll
        for (int q = 0; q < 4; ++q) {
            int cidx = l + 32 * q;
            h1[m * 128 + cidx] = gelu_f((v[q] - mean) * inv * g1[cidx] + bb1[cidx]);
        }
    }
    __syncthreads();
    // GEMM2: (16x128) @ (128x128)
    for (int nt = 0; nt < 8; ++nt) {
        v8f c = {};
        const int col = nt * 16 + mr;
        #pragma unroll
        for (int kb = 0; kb < 128; kb += 4) {
            v2f a, bf;
            a.x = h1[mr * 128 + kb + 2 * kh + 0];
            a.y = h1[mr * 128 + kb + 2 * kh + 1];
            bf.x = w2[col * 128 + kb + 2 * kh + 0];
            bf.y = w2[col * 128 + kb + 2 * kh + 1];
            c = __builtin_amdgcn_wmma_f32_16x16x4_f32(false, a, false, bf, (short)0, c, false, false);
        }
        #pragma unroll
        for (int r = 0; r < 8; ++r)
            sp[(r + 8 * kh) * 128 + col] = c[r] + b2[col];
    }
    __syncthreads();
    // LN, write spatial, fs2
    for (int m = 0; m < 16; ++m) {
        float v[4];
        #pragma unroll
        for (int q = 0; q < 4; ++q) v[q] = sp[m * 128 + l + 32 * q];
        float mean = warp_sum(v[0] + v[1] + v[2] + v[3]) * (1.0f / 128.0f);
        float sq = 0.f;
        #pragma unroll
        for (int q = 0; q < 4; ++q) { float d = v[q] - mean; sq += d * d; }
        float inv = rsqrtf(warp_sum(sq) * (1.0f / 128.0f) + 1e-5f);
        float ss = 0.f;
        #pragma unroll
        for (int q = 0; q < 4; ++q) {
            int cidx = l + 32 * q;
            float o = (v[q] - mean) * inv * g2[cidx] + bb2[cidx];
            spatial[(size_t)(tok0 + m) * 128 + cidx] = o;
            ss += o * o;
        }
        float tot = warp_sum(ss);
        if (l == 0) fs2[tok0 + m] = tot;
    }
}

// ---------------------------------------------------------------------------
// Kernel 2: local density.  Block = (batch, 16-row tile), 4 waves.
// dist tiles via f32 WMMA; per-lane running 6-smallest; merge; tanh(mean of 5 NN).
// ---------------------------------------------------------------------------
__global__ void density_kernel(const float* __restrict__ x, float* __restrict__ density) {
    __shared__ float xr[16 * 64];
    __shared__ float x2c[1024];
    __shared__ float stag[4][16][17];
    __shared__ float cand[16][8][6];
    const int b = blockIdx.x >> 6;
    const int row0 = (blockIdx.x & 63) * 16;
    const float* xb = x + (size_t)b * NN * 64;
    const int t = threadIdx.x;

    for (int i = t; i < 1024; i += 128) {
        const float* p = xb + (size_t)i * 64;
        float s = 0.f;
        #pragma unroll 8
        for (int k = 0; k < 64; ++k) { float v = p[k]; s += v * v; }
        x2c[i] = s;
    }
    for (int i = t; i < 16 * 64; i += 128)
        xr[i] = xb[(size_t)(row0 + (i >> 6)) * 64 + (i & 63)];
    __syncthreads();

    const int w = t >> 5, l = t & 31, mr = l & 15, kh = l >> 4;
    float best[6];
    #pragma unroll
    for (int q = 0; q < 6; ++q) best[q] = 3.0e38f;

    for (int ct = w; ct < 64; ct += 4) {
        const int c0 = ct * 16;
        v8f c = {};
        #pragma unroll
        for (int kb = 0; kb < 64; kb += 4) {
            v2f a, bf;
            a.x = xr[mr * 64 + kb + 2 * kh + 0];
            a.y = xr[mr * 64 + kb + 2 * kh + 1];
            const float* pc = xb + (size_t)(c0 + mr) * 64 + kb + 2 * kh;
            bf.x = pc[0];
            bf.y = pc[1];
            c = __builtin_amdgcn_wmma_f32_16x16x4_f32(false, a, false, bf, (short)0, c, false, false);
        }
        #pragma unroll
        for (int r = 0; r < 8; ++r) stag[w][r + 8 * kh][mr] = c[r];
        __syncthreads();
        const float x2row = x2c[row0 + mr];
        #pragma unroll
        for (int j = 0; j < 8; ++j) {
            const int cc = kh * 8 + j;
            float d2 = x2row + x2c[c0 + cc] - 2.0f * stag[w][mr][cc];
            float dd = sqrtf(fmaxf(d2, 0.0f));
            if (dd < best[5]) {
                best[5] = dd;
                #pragma unroll
                for (int q = 5; q > 0; --q)
                    if (best[q] < best[q - 1]) { float tmp = best[q]; best[q] = best[q - 1]; best[q - 1] = tmp; }
            }
        }
        __syncthreads();
    }
    #pragma unroll
    for (int q = 0; q < 6; ++q) cand[mr][w * 2 + kh][q] = best[q];
    __syncthreads();
    if (t < 16) {
        float m6[6];
        #pragma unroll
        for (int q = 0; q < 6; ++q) m6[q] = 3.0e38f;
        for (int s = 0; s < 8; ++s)
            for (int q2 = 0; q2 < 6; ++q2) {
                float dd = cand[t][s][q2];
                if (dd < m6[5]) {
                    m6[5] = dd;
                    for (int q = 5; q > 0; --q)
                        if (m6[q] < m6[q - 1]) { float tmp = m6[q]; m6[q] = m6[q - 1]; m6[q - 1] = tmp; }
                }
            }
        // m6[0] is the self distance (0); mean of the next 5
        float mean = (m6[1] + m6[2] + m6[3] + m6[4] + m6[5]) * 0.2f;
        density[(size_t)b * NN + row0 + t] = tanhf(mean);
    }
}

// ---------------------------------------------------------------------------
// Kernel 3: scale predictor -> pred_time.  16 tokens per 1-wave block.
// ---------------------------------------------------------------------------
__global__ void sp_kernel(const float* __restrict__ spatial, const float* __restrict__ density,
                          const float* __restrict__ w1, const float* __restrict__ b1,
                          const float* __restrict__ g, const float* __restrict__ bb,
                          const float* __restrict__ w2, const float* __restrict__ b2,
                          float* __restrict__ pred_time) {
    __shared__ float spt[16 * 128];
    __shared__ float s1[16 * 128];
    __shared__ float dloc[16];
    const int tok0 = blockIdx.x * 16;
    const int l = threadIdx.x;
    for (int i = l; i < 16 * 128; i += 32)
        spt[i] = spatial[(size_t)(tok0 + (i >> 7)) * 128 + (i & 127)];
    if (l < 16) dloc[l] = density[tok0 + l];
    __syncthreads();

    const int mr = l & 15, kh = l >> 4;
    for (int nt = 0; nt < 8; ++nt) {
        v8f c = {};
        const int col = nt * 16 + mr;
        #pragma unroll
        for (int kb = 0; kb < 128; kb += 4) {
            v2f a, bf;
            a.x = spt[mr * 128 + kb + 2 * kh + 0];
            a.y = spt[mr * 128 + kb + 2 * kh + 1];
            bf.x = w1[col * 129 + kb + 2 * kh + 0];
            bf.y = w1[col * 129 + kb + 2 * kh + 1];
            c = __builtin_amdgcn_wmma_f32_16x16x4_f32(false, a, false, bf, (short)0, c, false, false);
        }
        const float wlast = w1[col * 129 + 128];
        #pragma unroll
        for (int r = 0; r < 8; ++r) {
            const int row = r + 8 * kh;
            s1[row * 128 + col] = c[r] + dloc[row] * wlast + b1[col];
        }
    }
    __syncthreads();
    // LN + GELU
    for (int m = 0; m < 16; ++m) {
        float v[4];
        #pragma unroll
        for (int q = 0; q < 4; ++q) v[q] = s1[m * 128 + l + 32 * q];
        float mean = warp_sum(v[0] + v[1] + v[2] + v[3]) * (1.0f / 128.0f);
        float sq = 0.f;
        #pragma unroll
        for (int q = 0; q < 4; ++q) { float d = v[q] - mean; sq += d * d; }
        float inv = rsqrtf(warp_sum(sq) * (1.0f / 128.0f) + 1e-5f);
        #pragma unroll
        for (int q = 0; q < 4; ++q) {
            int cidx = l + 32 * q;
            s1[m * 128 + cidx] = gelu_f((v[q] - mean) * inv * g[cidx] + bb[cidx]);
        }
    }
    __syncthreads();
    // 128 -> 1 head, softplus, pred_time
    for (int m = 0; m < 16; ++m) {
        float part = 0.f;
        #pragma unroll
        for (int q = 0; q < 4; ++q) part += s1[m * 128 + l + 32 * q] * w2[l + 32 * q];
        float tot = warp_sum(part);
        if (l == 0) {
            float z = tot + b2[0];
            float spl = fmaxf(z, 0.f) + log1pf(expf(-fabsf(z)));
            pred_time[tok0 + m] = 5.0f - 1.5f * dloc[m] + 0.5f * spl;
        }
    }
}

// ---------------------------------------------------------------------------
// Init slots: obj[b,o,:] = object_space[o,:], ss2[b,o] = |obj|^2
// ---------------------------------------------------------------------------
__global__ void init_obj_kernel(const float* __restrict__ object_space,
                                float* __restrict__ obj, float* __restrict__ ss2) {
    __shared__ float red[128];
    const int b = blockIdx.x >> 3, o = blockIdx.x & 7, t = threadIdx.x;
    float v = object_space[o * 128 + t];
    obj[(size_t)(b * 8 + o) * 128 + t] = v;
    red[t] = v * v;
    __syncthreads();
    for (int s = 64; s > 0; s >>= 1) { if (t < s) red[t] += red[t + s]; __syncthreads(); }
    if (t == 0) ss2[b * 8 + o] = red[0];
}

// ---------------------------------------------------------------------------
// dot_obj[b,o,n] = <obj[b,o,:], spatial[b,n,:]>.  Block = (b, 128-n chunk), 256 thr.
// ---------------------------------------------------------------------------
__global__ void dot_kernel(const float* __restrict__ spatial, const float* __restrict__ obj,
                           float* __restrict__ dotobj) {
    __shared__ float ob[8 * 128];
    const int b = blockIdx.x >> 3, ch = blockIdx.x & 7, t = threadIdx.x;
    for (int i = t; i < 1024; i += 256) ob[i] = obj[(size_t)b * 1024 + i];
    __syncthreads();
    const int n = ch * 128 + (t >> 1);
    const int oh = (t & 1) * 4;
    const float* sp = spatial + ((size_t)b * NN + n) * 128;
    float acc[4] = {0.f, 0.f, 0.f, 0.f};
    for (int h = 0; h < 128; ++h) {
        float s = sp[h];
        #pragma unroll
        for (int q = 0; q < 4; ++q) acc[q] += s * ob[(oh + q) * 128 + h];
    }
    #pragma unroll
    for (int q = 0; q < 4; ++q)
        dotobj[((size_t)b * 8 + oh + q) * NN + n] = acc[q];
}

// ---------------------------------------------------------------------------
// Scores + softmax over K=24.  Block = (b, 128-n chunk), 128 thr (one n each).
// ---------------------------------------------------------------------------
__global__ void attn_kernel(const float* __restrict__ dotobj, const float* __restrict__ ss2,
                            const float* __restrict__ fs2, const float* __restrict__ pred_time,
                            const float* __restrict__ density, const float* __restrict__ hscale,
                            float* __restrict__ attn) {
    __shared__ float ss[8];
    const int b = blockIdx.x >> 3, ch = blockIdx.x & 7, t = threadIdx.x;
    if (t < 8) ss[t] = ss2[b * 8 + t];
    __syncthreads();
    const int n = ch * 128 + t;
    const float ft = pred_time[(size_t)b * NN + n];
    const float dn = density[(size_t)b * NN + n];
    const float f2 = fs2[(size_t)b * NN + n];
    const float mod = hscale[0] * (dn - 0.5f);
    const float LT[3] = {1.0f, 2.5f, 4.0f};
    const float BH[3] = {0.9f, 0.6f, 0.3f};
    float sc[24];
    float mx = -3.0e38f;
    for (int o = 0; o < 8; ++o) {
        float dp = dotobj[((size_t)b * 8 + o) * NN + n];
        float r2 = fmaxf(f2 + ss[o] - 2.0f * dp, 0.0f);
        float r = sqrtf(r2 + 1e-12f);
        #pragma unroll
        for (int lv = 0; lv < 3; ++lv) {
            float dt = ft - LT[lv];
            float lsq = dt * dt - r2;
            float dsc = -sqrtf(fabsf(lsq) + 1e-6f);
            float adt = fabsf(dt);
            float hor = fminf(fmaxf(BH[lv] + mod, 0.1f), 1.0f);
            float cone = hor - r / (adt + 1e-6f) - 10.0f * fmaxf(-dt, 0.f) - 5.0f * fmaxf(r - adt, 0.f);
            float s = (dsc + 0.5f * tanhf(cone)) * 10.0f; // / TAU(0.1)
            sc[o * 3 + lv] = s;
            mx = fmaxf(mx, s);
        }
    }
    float sum = 0.f;
    #pragma unroll
    for (int k = 0; k < 24; ++k) { float e = expf(sc[k] - mx); sc[k] = e; sum += e; }
    float inv = 1.0f / sum;
    #pragma unroll
    for (int k = 0; k < 24; ++k)
        attn[((size_t)b * KK + k) * NN + n] = sc[k] * inv;
}

// ---------------------------------------------------------------------------
// S[b,k] = sum_n attn[b,k,n] + 1e-8
// ---------------------------------------------------------------------------
__global__ void colsum_kernel(const float* __restrict__ attn, float* __restrict__ S) {
    __shared__ float red[256];
    const int bk = blockIdx.x, t = threadIdx.x;
    const float* p = attn + (size_t)bk * NN;
    float s = 0.f;
    for (int i = t; i < 1024; i += 256) s += p[i];
    red[t] = s;
    __syncthreads();
    for (int st = 128; st > 0; st >>= 1) { if (t < st) red[t] += red[t + st]; __syncthreads(); }
    if (t == 0) S[bk] = red[0] + 1e-8f;
}

// ---------------------------------------------------------------------------
// Fused: weighted update + GRU cell + residual MLP -> new obj, ss2.
// Block = (b, o), 128 threads (one hidden dim each).
// ---------------------------------------------------------------------------
__global__ void update_kernel(const float* __restrict__ attn, const float* __restrict__ S,
                              const float* __restrict__ spatial, const float* __restrict__ object_space,
                              const float* __restrict__ gwih, const float* __restrict__ gwhh,
                              const float* __restrict__ gbih, const float* __restrict__ gbhh,
                              const float* __restrict__ mw1, const float* __restrict__ mb1,
                              const float* __restrict__ mw2, const float* __restrict__ mb2,
                              const float* __restrict__ lng, const float* __restrict__ lnb,
                              float* __restrict__ obj, float* __restrict__ ss2) {
    __shared__ float wbuf[128], ubuf[128], h0b[128], red[128], abuf[128], m1b[128];
    const int b = blockIdx.x >> 3, o = blockIdx.x & 7, t = threadIdx.x;
    const float inv0 = 1.0f / S[b * KK + 3 * o + 0];
    const float inv1 = 1.0f / S[b * KK + 3 * o + 1];
    const float inv2 = 1.0f / S[b * KK + 3 * o + 2];
    const float* a0 = attn + ((size_t)b * KK + 3 * o) * NN;
    float acc = 0.f;
    for (int nb = 0; nb < 8; ++nb) {
        const int n = nb * 128 + t;
        wbuf[t] = a0[n] * inv0 + a0[NN + n] * inv1 + a0[2 * NN + n] * inv2;
        __syncthreads();
        const float* spb = spatial + ((size_t)b * NN + nb * 128) * 128;
        #pragma unroll 4
        for (int j = 0; j < 128; ++j) acc += wbuf[j] * spb[(size_t)j * 128 + t];
        __syncthreads();
    }
    ubuf[t] = acc;
    const float h0 = object_space[o * 128 + t];
    h0b[t] = h0;
    __syncthreads();
    float gi[3], gh[3];
    #pragma unroll
    for (int p = 0; p < 3; ++p) {
        const int row = p * 128 + t;
        const float* wi = gwih + (size_t)row * 128;
        const float* wh = gwhh + (size_t)row * 128;
        float ai = 0.f, ah = 0.f;
        for (int j = 0; j < 128; ++j) { ai += wi[j] * ubuf[j]; ah += wh[j] * h0b[j]; }
        gi[p] = ai + gbih[row];
        gh[p] = ah + gbhh[row];
    }
    const float rg = sigmoid_f(gi[0] + gh[0]);
    const float zg = sigmoid_f(gi[1] + gh[1]);
    const float ng = tanhf(gi[2] + rg * gh[2]);
    const float nw = (1.0f - zg) * ng + zg * h0;
    // LayerNorm over the 128 dims
    red[t] = nw; __syncthreads();
    for (int s = 64; s > 0; s >>= 1) { if (t < s) red[t] += red[t + s]; __syncthreads(); }
    const float mean = red[0] * (1.0f / 128.0f);
    __syncthreads();
    const float d = nw - mean;
    red[t] = d * d; __syncthreads();
    for (int s = 64; s > 0; s >>= 1) { if (t < s) red[t] += red[t + s]; __syncthreads(); }
    const float var = red[0] * (1.0f / 128.0f);
    __syncthreads();
    abuf[t] = d * rsqrtf(var + 1e-5f) * lng[t] + lnb[t];
    __syncthreads();
    const float* w1p = mw1 + (size_t)t * 128;
    float m1 = 0.f;
    for (int j = 0; j < 128; ++j) m1 += w1p[j] * abuf[j];
    m1b[t] = gelu_f(m1 + mb1[t]);
    __syncthreads();
    const float* w2p = mw2 + (size_t)t * 128;
    float m2 = 0.f;
    for (int j = 0; j < 128; ++j) m2 += w2p[j] * m1b[j];
    const float y = nw + 0.2f * (m2 + mb2[t]);
    obj[((size_t)b * 8 + o) * 128 + t] = y;
    red[t] = y * y; __syncthreads();
    for (int s = 64; s > 0; s >>= 1) { if (t < s) red[t] += red[t + s]; __syncthreads(); }
    if (t == 0) ss2[b * 8 + o] = red[0];
}

// ---------------------------------------------------------------------------
// slots[b,k,:] = [time_k, obj[b, k/3, :]]
// ---------------------------------------------------------------------------
__global__ void slots_kernel(const float* __restrict__ obj, float* __restrict__ slots) {
    const int bk = blockIdx.x, b = bk / 24, k = bk % 24, t = threadIdx.x;
    const float LT[3] = {1.0f, 2.5f, 4.0f};
    float* row = slots + (size_t)bk * 129;
    if (t == 0) row[0] = LT[k % 3];
    if (t >= 1 && t < 129)
        row[t] = obj[((size_t)b * 8 + k / 3) * 128 + (t - 1)];
}

// ---------------------------------------------------------------------------
extern "C" void kernel_launch(void* const* d_in, const int* in_sizes, int n_in,
                              void* d_out, int out_size, void* d_ws, size_t ws_size,
                              hipStream_t stream) {
    (void)in_sizes; (void)n_in; (void)out_size; (void)ws_size;
    const float* x        = (const float*)d_in[0];
    const float* fe_w1    = (const float*)d_in[1];
    const float* fe_b1    = (const float*)d_in[2];
    const float* fe_g1    = (const float*)d_in[3];
    const float* fe_bb1   = (const float*)d_in[4];
    const float* fe_w2    = (const float*)d_in[5];
    const float* fe_b2    = (const float*)d_in[6];
    const float* fe_g2    = (const float*)d_in[7];
    const float* fe_bb2   = (const float*)d_in[8];
    const float* sp_w1    = (const float*)d_in[9];
    const float* sp_b1    = (const float*)d_in[10];
    const float* sp_g     = (const float*)d_in[11];
    const float* sp_bb    = (const float*)d_in[12];
    const float* sp_w2    = (const float*)d_in[13];
    const float* sp_b2    = (const float*)d_in[14];
    const float* obj_sp   = (const float*)d_in[15];
    const float* hscale   = (const float*)d_in[16];
    const float* gw_ih    = (const float*)d_in[17];
    const float* gw_hh    = (const float*)d_in[18];
    const float* gb_ih    = (const float*)d_in[19];
    const float* gb_hh    = (const float*)d_in[20];
    const float* mlp_w1   = (const float*)d_in[21];
    const float* mlp_b1   = (const float*)d_in[22];
    const float* mlp_w2   = (const float*)d_in[23];
    const float* mlp_b2   = (const float*)d_in[24];
    const float* ln_g     = (const float*)d_in[25];
    const float* ln_b     = (const float*)d_in[26];

    float* out = (float*)d_out;
    // output layout: slots(32*24*129) | attn(32*24*1024) | pred_time(32*1024) | density(32*1024)
    float* out_slots = out;
    float* out_attn  = out + 99072;
    float* out_pt    = out + 99072 + 786432;
    float* out_den   = out + 99072 + 786432 + 32768;

    float* ws = (float*)d_ws;
    float* ws_spatial = ws;                       // 32*1024*128
    float* ws_fs2     = ws_spatial + 4194304;     // 32*1024
    float* ws_dot     = ws_fs2 + 32768;           // 32*8*1024
    float* ws_obj     = ws_dot + 262144;          // 32*8*128
    float* ws_ss2     = ws_obj + 32768;           // 32*8
    float* ws_S       = ws_ss2 + 256;             // 32*24

    const int BN16 = (BB * NN) / 16;              // 2048

    fe_kernel<<<BN16, 32, 0, stream>>>(x, fe_w1, fe_b1, fe_g1, fe_bb1,
                                       fe_w2, fe_b2, fe_g2, fe_bb2,
                                       ws_spatial, ws_fs2);
    density_kernel<<<BB * (NN / 16), 128, 0, stream>>>(x, out_den);
    sp_kernel<<<BN16, 32, 0, stream>>>(ws_spatial, out_den, sp_w1, sp_b1, sp_g, sp_bb,
                                       sp_w2, sp_b2, out_pt);
    init_obj_kernel<<<BB * OO, 128, 0, stream>>>(obj_sp, ws_obj, ws_ss2);

    for (int it = 0; it < 3; ++it) {
        dot_kernel<<<BB * 8, 256, 0, stream>>>(ws_spatial, ws_obj, ws_dot);
        attn_kernel<<<BB * 8, 128, 0, stream>>>(ws_dot, ws_ss2, ws_fs2, out_pt, out_den,
                                                hscale, out_attn);
        colsum_kernel<<<BB * KK, 256, 0, stream>>>(out_attn, ws_S);
        update_kernel<<<BB * OO, 128, 0, stream>>>(out_attn, ws_S, ws_spatial, obj_sp,
                                                   gw_ih, gw_hh, gb_ih, gb_hh,
                                                   mlp_w1, mlp_b1, mlp_w2, mlp_b2,
                                                   ln_g, ln_b, ws_obj, ws_ss2);
    }
    slots_kernel<<<BB * KK, 160, 0, stream>>>(ws_obj, out_slots);
}